// GaussianAttention_11364483465393
// MI455X (gfx1250) — compile-verified
//
#include <hip/hip_runtime.h>

typedef __attribute__((ext_vector_type(2))) float v2f;
typedef __attribute__((ext_vector_type(4))) float v4f;
typedef __attribute__((ext_vector_type(8))) float v8f;

#define B_   64
#define T_   512
#define H_   1024
#define K_   10
#define U_   128
#define C_   128
#define OUTW 1155   /* H + C + 3 */

// ---------------------------------------------------------------------------
// prep 0: pair-swizzle window_w (H x 30) -> Wsw[H/2][64]:
//         Wsw[p][2n+q] = (n<30) ? W[2p+q][n] : 0      (K pairs contiguous)
// ---------------------------------------------------------------------------
__global__ void k_swz_w(const float* __restrict__ w, float* __restrict__ wsw) {
    int i = blockIdx.x * blockDim.x + threadIdx.x;
    if (i >= (H_ / 2) * 64) return;
    int p = i >> 6;
    int t = i & 63;
    int n = t >> 1;
    int q = t & 1;
    wsw[i] = (n < 30) ? w[(2 * p + q) * 30 + n] : 0.0f;
}

// ---------------------------------------------------------------------------
// prep 1: pair-swizzle char_seq (B x U x C) -> csw[B][U/2][2*C]:
//         csw[b][p][2n+q] = cs[b][2p+q][n]
// ---------------------------------------------------------------------------
__global__ void k_swz_cs(const float* __restrict__ cs, float* __restrict__ csw) {
    int i = blockIdx.x * blockDim.x + threadIdx.x;
    if (i >= B_ * (U_ / 2) * (2 * C_)) return;
    int b   = i >> 14;           // /(64*256)
    int rem = i & 16383;
    int p   = rem >> 8;
    int t   = rem & 255;
    int n   = t >> 1;
    int q   = t & 1;
    csw[i] = cs[((size_t)b * U_ + 2 * p + q) * C_ + n];
}

// ---------------------------------------------------------------------------
// copy `original` into out[:, :, 1152:1155]  (write-once -> nontemporal)
// ---------------------------------------------------------------------------
__global__ void k_copy_orig(const float* __restrict__ orig, float* __restrict__ out) {
    int i = blockIdx.x * blockDim.x + threadIdx.x;
    if (i >= B_ * T_ * 3) return;
    int row = i / 3, j = i - row * 3;
    __builtin_nontemporal_store(orig[i], &out[(size_t)row * OUTW + H_ + C_ + j]);
}

// ---------------------------------------------------------------------------
// kernel 1: abk = exp(X @ W + b) via V_WMMA_F32_16X16X4_F32
//           + fused copy of X into out[:, :, 0:1024] (X streamed exactly once)
// grid 512 x 128 threads (4 waves); block handles 64 rows; wave handles 16.
// ---------------------------------------------------------------------------
__global__ __launch_bounds__(128) void k_abk(const float* __restrict__ x,
                                             const float* __restrict__ wsw,
                                             const float* __restrict__ bias,
                                             float* __restrict__ abk,
                                             float* __restrict__ out) {
    __shared__ float As[64 * 68];   // A tile, stride 68: bank=(4*row+k)%64, conflict-free
    __shared__ float Bs[32 * 96];   // 32 K-pair rows; stride 96 == 32 mod 64 banks ->
                                    // upper half-wave (pair-row p+1) uses the other 32 banks

    const int tid  = threadIdx.x;
    const int lane = tid & 31;
    const int wv   = tid >> 5;
    const int lo   = lane & 15;
    const int hi   = lane >> 4;
    const int row0 = blockIdx.x * 64;

    v8f acc0 = {0.f,0.f,0.f,0.f,0.f,0.f,0.f,0.f};
    v8f acc1 = {0.f,0.f,0.f,0.f,0.f,0.f,0.f,0.f};

    for (int k0 = 0; k0 < H_; k0 += 64) {
        // stage A (64 rows x 64 cols) coalesced, fuse the out-copy of X
        #pragma unroll
        for (int i = 0; i < 8; ++i) {
            int idx = i * 128 + tid;
            int r   = idx >> 4;
            int c4  = (idx & 15) << 2;
            const v4f v = __builtin_nontemporal_load(
                (const v4f*)(x + (size_t)(row0 + r) * H_ + k0 + c4));
            *(v4f*)(&As[r * 68 + c4]) = v;
            float* o = out + (size_t)(row0 + r) * OUTW + k0 + c4;
            __builtin_nontemporal_store(v.x, o + 0);
            __builtin_nontemporal_store(v.y, o + 1);
            __builtin_nontemporal_store(v.z, o + 2);
            __builtin_nontemporal_store(v.w, o + 3);
        }
        // stage B chunk: 32 K-pair rows x 64 (pair-interleaved) floats
        #pragma unroll
        for (int i = 0; i < 4; ++i) {
            int idx = i * 128 + tid;      // 0..511
            int r   = idx >> 4;           // pair row 0..31
            int c4  = (idx & 15) << 2;    // 0..60
            *(v4f*)(&Bs[r * 96 + c4]) =
                *(const v4f*)(wsw + (size_t)((k0 >> 1) + r) * 64 + c4);
        }
        __syncthreads();

        const int arow = (wv * 16 + lo) * 68;
        #pragma unroll
        for (int k = 0; k < 64; k += 4) {
            // A 16x4 f32: lanes0-15 K={0,1}, lanes16-31 K={2,3}
            v2f a = *(const v2f*)(&As[arow + k + hi * 2]);
            const int bro = ((k >> 1) + hi) * 96;
            v2f b0 = *(const v2f*)(&Bs[bro + lo * 2]);        // {B[k+2hi][lo],   B[k+2hi+1][lo]}
            v2f b1 = *(const v2f*)(&Bs[bro + 32 + lo * 2]);   // same, N = 16+lo
            acc0 = __builtin_amdgcn_wmma_f32_16x16x4_f32(false, a, false, b0,
                                                         (short)0, acc0, false, false);
            acc1 = __builtin_amdgcn_wmma_f32_16x16x4_f32(false, a, false, b1,
                                                         (short)0, acc1, false, false);
        }
        __syncthreads();
    }

    // epilogue: bias + exp, store abk padded to stride 32
    #pragma unroll
    for (int r = 0; r < 8; ++r) {
        int m    = wv * 16 + (hi ? r + 8 : r);   // C/D: VGPR r -> row r / r+8
        int grow = row0 + m;
        {
            int   col = lo;                       // < 16, always valid bias
            float bv  = bias[col];
            abk[(size_t)grow * 32 + col] = __expf(acc0[r] + bv);
        }
        {
            int   col = 16 + lo;
            float bv  = (col < 30) ? bias[col] : 0.0f;
            abk[(size_t)grow * 32 + col] = __expf(acc1[r] + bv);
        }
    }
}

// ---------------------------------------------------------------------------
// kernel 2: kappa = init_kappa + cumsum(kappa_inc, axis=T)
// one wave32 per (b,k): lane owns a 16-long chunk; wave shfl-scan of chunk sums
// ---------------------------------------------------------------------------
__global__ __launch_bounds__(32) void k_cumsum(const float* __restrict__ abk,
                                               const float* __restrict__ init_kappa,
                                               float* __restrict__ kap) {
    int b    = blockIdx.x / K_;
    int k    = blockIdx.x - b * K_;
    int lane = threadIdx.x;
    int t0   = lane * 16;

    float inc[16];
    float s = 0.f;
    #pragma unroll
    for (int i = 0; i < 16; ++i) {
        inc[i] = abk[(size_t)(b * T_ + t0 + i) * 32 + 20 + k];
        s += inc[i];
    }
    // inclusive wave32 scan of chunk sums
    float sc = s;
    #pragma unroll
    for (int d = 1; d < 32; d <<= 1) {
        float o = __shfl_up(sc, d, 32);
        if (lane >= d) sc += o;
    }
    float run = sc - s + init_kappa[b * K_ + k];   // exclusive prefix + init
    #pragma unroll
    for (int i = 0; i < 16; ++i) {
        run += inc[i];
        kap[(size_t)(b * T_ + t0 + i) * 16 + k] = run;
    }
}

// ---------------------------------------------------------------------------
// kernel 3: phi[t,u] = sum_k alpha*exp(-beta*(kappa-u)^2)  (into LDS),
//           window = phi @ char_seq via V_WMMA_F32_16X16X4_F32,
//           NT-store into out[:, :, 1024:1152].
// grid 512 (= 64 b x 8 t-tiles) x 128 threads; block: 64 t rows, full U,C=128.
// ---------------------------------------------------------------------------
__global__ __launch_bounds__(128) void k_window(const float* __restrict__ abk,
                                                const float* __restrict__ kap,
                                                const float* __restrict__ csw,
                                                float* __restrict__ out) {
    __shared__ float Ph[64 * 132];   // stride 132: bank=(4*t+k)%64 -> conflict-free A reads

    const int tid = threadIdx.x;
    const int b   = blockIdx.x >> 3;
    const int t0  = (blockIdx.x & 7) * 64;

    // --- phi: thread owns column u = tid, loops over the 64 t rows ---------
    const int   u  = tid;
    const float uf = (float)u;
    for (int t = 0; t < 64; ++t) {
        const int    grow = b * T_ + t0 + t;
        const float* arow = abk + (size_t)grow * 32;   // alpha[0..9], beta[10..19]
        const float* krow = kap + (size_t)grow * 16;
        float acc = 0.f;
        #pragma unroll
        for (int k = 0; k < K_; ++k) {
            float d = krow[k] - uf;                    // uniform-addr loads -> scalar
            acc += arow[k] * __expf(-arow[10 + k] * d * d);
        }
        Ph[t * 132 + u] = acc;
    }
    __syncthreads();

    // --- window GEMM: M=16 per wave, N=8 tiles of 16, K=U=128 --------------
    const int lane = tid & 31;
    const int wv   = tid >> 5;
    const int lo   = lane & 15;
    const int hi   = lane >> 4;
    const float* bp   = csw + (size_t)b * ((U_ / 2) * (2 * C_));  // pair-swizzled, L2-resident
    const int    arow = (wv * 16 + lo) * 132;

    #pragma unroll 1
    for (int j = 0; j < 8; ++j) {
        v8f acc = {0.f,0.f,0.f,0.f,0.f,0.f,0.f,0.f};
        #pragma unroll
        for (int k = 0; k < U_; k += 4) {
            v2f a  = *(const v2f*)(&Ph[arow + k + hi * 2]);
            // one coalesced b64: {cs[k+2hi][n], cs[k+2hi+1][n]}, n = j*16+lo
            v2f bb = *(const v2f*)(&bp[(size_t)((k >> 1) + hi) * 256 + (j * 16 + lo) * 2]);
            acc = __builtin_amdgcn_wmma_f32_16x16x4_f32(false, a, false, bb,
                                                        (short)0, acc, false, false);
        }
        #pragma unroll
        for (int r = 0; r < 8; ++r) {
            int t = t0 + wv * 16 + (hi ? r + 8 : r);
            __builtin_nontemporal_store(
                acc[r], &out[(size_t)(b * T_ + t) * OUTW + H_ + j * 16 + lo]);
        }
    }
}

// ---------------------------------------------------------------------------
extern "C" void kernel_launch(void* const* d_in, const int* in_sizes, int n_in,
                              void* d_out, int out_size, void* d_ws, size_t ws_size,
                              hipStream_t stream) {
    const float* input0     = (const float*)d_in[0];
    const float* original   = (const float*)d_in[1];
    const float* init_kappa = (const float*)d_in[2];
    const float* char_seq   = (const float*)d_in[3];
    const float* window_w   = (const float*)d_in[4];
    const float* window_b   = (const float*)d_in[5];
    float* out = (float*)d_out;

    char*  ws  = (char*)d_ws;
    float* wsw = (float*)(ws);                                 // 512*64*4    = 128 KB
    float* abk = (float*)(ws + (size_t)131072);                // 32768*32*4  = 4 MB
    float* kap = (float*)(ws + (size_t)131072 + 4194304);      // 32768*16*4  = 2 MB
    float* csw = (float*)(ws + (size_t)131072 + 4194304 + 2097152);  // 64*64*256*4 = 4 MB

    hipLaunchKernelGGL(k_swz_w,     dim3(128),  dim3(256), 0, stream, window_w, wsw);
    hipLaunchKernelGGL(k_swz_cs,    dim3(4096), dim3(256), 0, stream, char_seq, csw);
    hipLaunchKernelGGL(k_copy_orig, dim3(384),  dim3(256), 0, stream, original, out);
    hipLaunchKernelGGL(k_abk,       dim3(512),  dim3(128), 0, stream,
                       input0, wsw, window_b, abk, out);
    hipLaunchKernelGGL(k_cumsum,    dim3(640),  dim3(32),  0, stream,
                       abk, init_kappa, kap);
    hipLaunchKernelGGL(k_window,    dim3(512),  dim3(128), 0, stream,
                       abk, kap, csw, out);
}